// GetCostVolume_3410204033788
// MI455X (gfx1250) — compile-verified
//
#include <hip/hip_runtime.h>

// Problem constants (match reference shapes)
#define CG    320          // gwc channels
#define CC    32           // concat channels
#define Hh    56
#define Ww    112
#define ND    48
#define NG    40           // groups (cpg = 8)
#define HW    (Hh * Ww)    // 6272
#define DHW   (ND * HW)    // 301056
#define WT    7            // 16-pixel tiles per row (112/16)

typedef float v2f __attribute__((ext_vector_type(2)));
typedef float v4f __attribute__((ext_vector_type(4)));
typedef float v8f __attribute__((ext_vector_type(8)));

// ---------------------------------------------------------------------------
// LDS tiled transpose: (C, N) channel-major -> (N, C) pixel-major.
// Both reads and writes are fully coalesced. C % 32 == 0, N % 32 == 0.
// ---------------------------------------------------------------------------
__global__ __launch_bounds__(256) void xpose_kernel(const float* __restrict__ src,
                                                    float* __restrict__ dst,
                                                    int C, int N) {
  __shared__ float tile[32][33];
  const int n0 = blockIdx.x * 32;
  const int c0 = blockIdx.y * 32;
  const int tx = threadIdx.x;        // 0..31
  const int ty = threadIdx.y;        // 0..7
#pragma unroll
  for (int i = 0; i < 32; i += 8)
    tile[ty + i][tx] = src[(size_t)(c0 + ty + i) * N + (n0 + tx)];
  __syncthreads();
#pragma unroll
  for (int i = 0; i < 32; i += 8)
    dst[(size_t)(n0 + ty + i) * C + (c0 + tx)] = tile[tx][ty + i];
}

// ---------------------------------------------------------------------------
// 1-D bilinear warp parameters (matches reference masks/clamps exactly).
// ---------------------------------------------------------------------------
struct WarpP { float w0, w1, valid; int x0c, x1c; };

__device__ __forceinline__ WarpP warp_params(int w, float dsp) {
  WarpP r;
  float xs  = (float)w - dsp;
  float x0f = floorf(xs);
  float fr  = xs - x0f;
  int   x0  = (int)x0f;
  int   x1  = x0 + 1;
  r.w0   = (x0 >= 0 && x0 <= Ww - 1) ? (1.0f - fr) : 0.0f;
  r.w1   = (x1 >= 0 && x1 <= Ww - 1) ? fr : 0.0f;
  r.x0c  = x0 < 0 ? 0 : (x0 > Ww - 1 ? Ww - 1 : x0);
  r.x1c  = x1 < 0 ? 0 : (x1 > Ww - 1 ? Ww - 1 : x1);
  r.valid = ((float)w >= dsp) ? 1.0f : 0.0f;
  return r;
}

// ---------------------------------------------------------------------------
// Group-wise correlation volume via V_WMMA_F32_16X16X4_F32 (exact fp32).
// One wave handles one (d, h, 16-pixel tile). M=N=16 pixels, K=4 channels;
// diagonal of D = per-pixel group dot. Two chained WMMAs give the 8-channel
// group; 40 groups -> 80 WMMA issues per wave-tile.
//   lane layout (A and B identical): pixel p = lane%16, K = 2*(lane/16)+j.
// XP=true: X/Y are pixel-major (HW, CG) -> float2 channel loads.
// ---------------------------------------------------------------------------
template <bool XP>
__global__ __launch_bounds__(256) void gwc_wmma_kernel(const float* __restrict__ X,
                                                       const float* __restrict__ Y,
                                                       const float* __restrict__ disp,
                                                       float* __restrict__ out) {
  const int wid  = blockIdx.x * 8 + (threadIdx.x >> 5);   // wave id, exact grid
  const int lane = threadIdx.x & 31;
  const int d    = wid / (Hh * WT);
  const int rem  = wid % (Hh * WT);
  const int h    = rem / WT;
  const int wt   = rem % WT;

  const int p  = lane & 15;       // pixel within tile == matrix row/col
  const int kk = lane >> 4;       // selects K pair {2kk, 2kk+1}
  const int w  = wt * 16 + p;

  const float dsp = disp[(size_t)d * HW + h * Ww + w];
  const WarpP wp  = warp_params(w, dsp);
  const float scale = 0.125f * wp.valid;   // mean over cpg=8, folded with x-mask

  const int q  = h * Ww + w;       // own pixel
  const int q0 = h * Ww + wp.x0c;  // warped gather columns
  const int q1 = h * Ww + wp.x1c;

  auto ld2 = [&](const float* __restrict__ A, int qq, int c) -> v2f {
    if constexpr (XP) {
      return *(const v2f*)(A + (size_t)qq * CG + c);          // contiguous pair
    } else {
      v2f r;
      r.x = A[(size_t)c * HW + qq];
      r.y = A[(size_t)(c + 1) * HW + qq];
      return r;
    }
  };

  const bool  act = (lane < 8) || (lane >= 24);   // lanes holding the diagonal
  const int   m8  = lane & 7;                     // accumulator index to pick
  const size_t obase = (size_t)d * HW + h * Ww + w;

#pragma unroll 4
  for (int g = 0; g < NG; ++g) {
    const int cbase = g * 8 + kk * 2;
    v8f acc = {0.f, 0.f, 0.f, 0.f, 0.f, 0.f, 0.f, 0.f};

    // K-chunk 0: channels cbase..cbase+1
    v2f a0 = ld2(X, q,  cbase);
    v2f y00 = ld2(Y, q0, cbase);
    v2f y01 = ld2(Y, q1, cbase);
    v2f b0; b0.x = wp.w0 * y00.x + wp.w1 * y01.x;
            b0.y = wp.w0 * y00.y + wp.w1 * y01.y;
    acc = __builtin_amdgcn_wmma_f32_16x16x4_f32(false, a0, false, b0,
                                                (short)0, acc, false, false);

    // K-chunk 1: channels cbase+4..cbase+5 (accumulate into same C)
    v2f a1 = ld2(X, q,  cbase + 4);
    v2f y10 = ld2(Y, q0, cbase + 4);
    v2f y11 = ld2(Y, q1, cbase + 4);
    v2f b1; b1.x = wp.w0 * y10.x + wp.w1 * y11.x;
            b1.y = wp.w0 * y10.y + wp.w1 * y11.y;
    acc = __builtin_amdgcn_wmma_f32_16x16x4_f32(false, a1, false, b1,
                                                (short)0, acc, false, false);

    // Diagonal (m,m): m<8 -> lane m, VGPR m;  m>=8 -> lane m+16, VGPR m-8.
    float dv = acc[0];
    dv = (m8 == 1) ? acc[1] : dv;
    dv = (m8 == 2) ? acc[2] : dv;
    dv = (m8 == 3) ? acc[3] : dv;
    dv = (m8 == 4) ? acc[4] : dv;
    dv = (m8 == 5) ? acc[5] : dv;
    dv = (m8 == 6) ? acc[6] : dv;
    dv = (m8 == 7) ? acc[7] : dv;
    if (act) out[(size_t)g * DHW + obase] = dv * scale;   // own-pixel store
  }
}

// ---------------------------------------------------------------------------
// Concat volume: channels [NG, NG+CC) = x_concat * valid,
//                channels [NG+CC, NG+2*CC) = warped y_concat.
// Thread per (d,h,w): w fastest -> coalesced stores per channel.
// ---------------------------------------------------------------------------
template <bool XP>
__global__ __launch_bounds__(256) void concat_kernel(const float* __restrict__ XC,
                                                     const float* __restrict__ YC,
                                                     const float* __restrict__ disp,
                                                     float* __restrict__ out) {
  const int idx = blockIdx.x * 256 + threadIdx.x;  // d*HW + h*W + w, exact grid
  const int w   = idx % Ww;
  const int h   = (idx / Ww) % Hh;

  const float dsp = disp[idx];
  const WarpP wp  = warp_params(w, dsp);

  const int q  = h * Ww + w;
  const int q0 = h * Ww + wp.x0c;
  const int q1 = h * Ww + wp.x1c;

  auto ld4 = [&](const float* __restrict__ A, int qq, int c) -> v4f {
    if constexpr (XP) {
      return *(const v4f*)(A + (size_t)qq * CC + c);
    } else {
      v4f r;
      r.x = A[(size_t)(c + 0) * HW + qq];
      r.y = A[(size_t)(c + 1) * HW + qq];
      r.z = A[(size_t)(c + 2) * HW + qq];
      r.w = A[(size_t)(c + 3) * HW + qq];
      return r;
    }
  };

#pragma unroll
  for (int c = 0; c < CC; c += 4) {
    v4f xv = ld4(XC, q,  c);
    v4f a0 = ld4(YC, q0, c);
    v4f a1 = ld4(YC, q1, c);
#pragma unroll
    for (int j = 0; j < 4; ++j) {
      out[(size_t)(NG + c + j) * DHW + idx]      = xv[j] * wp.valid;
      out[(size_t)(NG + CC + c + j) * DHW + idx] = wp.w0 * a0[j] + wp.w1 * a1[j];
    }
  }
}

// ---------------------------------------------------------------------------
extern "C" void kernel_launch(void* const* d_in, const int* in_sizes, int n_in,
                              void* d_out, int out_size, void* d_ws, size_t ws_size,
                              hipStream_t stream) {
  (void)in_sizes; (void)n_in; (void)out_size;
  const float* x_gwc = (const float*)d_in[0];
  const float* y_gwc = (const float*)d_in[1];
  const float* x_cat = (const float*)d_in[2];
  const float* y_cat = (const float*)d_in[3];
  const float* disp  = (const float*)d_in[4];
  float* out = (float*)d_out;

  const size_t bytes_g = (size_t)CG * HW * sizeof(float);   // 8.03 MB each
  const size_t bytes_c = (size_t)CC * HW * sizeof(float);   // 0.80 MB each
  const size_t need    = 2 * bytes_g + 2 * bytes_c;         // ~17.7 MB

  const dim3 tb(32, 8);
  const int gwc_blocks    = (ND * Hh * WT) / 8;   // 18816 waves / 8 per block
  const int concat_blocks = DHW / 256;            // exact

  if (ws_size >= need) {
    char* ws = (char*)d_ws;
    float* xgT = (float*)(ws);
    float* ygT = (float*)(ws + bytes_g);
    float* xcT = (float*)(ws + 2 * bytes_g);
    float* ycT = (float*)(ws + 2 * bytes_g + bytes_c);

    // Pixel-major copies: turn 25KB-strided channel walks into b64/b128 loads.
    xpose_kernel<<<dim3(HW / 32, CG / 32), tb, 0, stream>>>(x_gwc, xgT, CG, HW);
    xpose_kernel<<<dim3(HW / 32, CG / 32), tb, 0, stream>>>(y_gwc, ygT, CG, HW);
    xpose_kernel<<<dim3(HW / 32, CC / 32), tb, 0, stream>>>(x_cat, xcT, CC, HW);
    xpose_kernel<<<dim3(HW / 32, CC / 32), tb, 0, stream>>>(y_cat, ycT, CC, HW);

    gwc_wmma_kernel<true><<<gwc_blocks, 256, 0, stream>>>(xgT, ygT, disp, out);
    concat_kernel<true><<<concat_blocks, 256, 0, stream>>>(xcT, ycT, disp, out);
  } else {
    // Workspace too small: same math on original channel-major layout.
    gwc_wmma_kernel<false><<<gwc_blocks, 256, 0, stream>>>(x_gwc, y_gwc, disp, out);
    concat_kernel<false><<<concat_blocks, 256, 0, stream>>>(x_cat, y_cat, disp, out);
  }
}